// Decoder_69071664054910
// MI455X (gfx1250) — compile-verified
//
#include <hip/hip_runtime.h>
#include <hip/hip_bf16.h>

typedef __attribute__((ext_vector_type(2))) float v2f;
typedef __attribute__((ext_vector_type(4))) float v4f;
typedef __attribute__((ext_vector_type(8))) float v8f;

#define BATCH 64
#define SLEN  128
#define HID   512
#define EMB   512
#define VOCAB 32000

#define WMMA_F32(a, b, c) \
    __builtin_amdgcn_wmma_f32_16x16x4_f32(false, (a), false, (b), (short)0, (c), false, false)

static __device__ __forceinline__ v2f lo2(v4f v) { return __builtin_shufflevector(v, v, 0, 1); }
static __device__ __forceinline__ v2f hi2(v4f v) { return __builtin_shufflevector(v, v, 2, 3); }

// ---------------------------------------------------------------------------
// Generic fp32 WMMA GEMM:  D[64, N] = X[64, K] @ W[N, K]^T + bias[N]
// One wave owns TWO 16-wide N tiles (32 columns) and all four 16-row M tiles
// (M = 64): every W element streams from HBM exactly once (NT hint, full M
// reuse), and the A-operand (X, L2-resident) loads are amortized over both
// N tiles, halving L2 read traffic vs. one tile per wave.
//
// Inner loop handles 8 k-values per iteration with ONE b128 load per operand
// row per lane (at k0 + 4*half): lane-half h supplies the k-pair slots
// (2h, 2h+1) identically for A and B, so the low float-pair of the b128
// feeds WMMA step 1 and the high pair feeds step 2 — a k-permutation that
// leaves the dot product invariant.  Register double-buffering issues
// next-chunk loads before current-chunk WMMAs so the load pipe never drains.
// Requires: K % 16 == 0, N % 32 == 0.
// ---------------------------------------------------------------------------
__global__ void wmma_gemm_xwT(const float* __restrict__ X,
                              const float* __restrict__ W,
                              const float* __restrict__ bias,
                              float* __restrict__ D,
                              int N, int K) {
    const int lane = threadIdx.x & 31;
    const int wave = threadIdx.x >> 5;                 // wave32
    const int unit = blockIdx.x * (blockDim.x >> 5) + wave;
    const int nunits = N >> 5;                         // 32 columns per wave
    if (unit >= nunits) return;                        // wave-uniform: EXEC stays full
    const int n0   = unit << 5;
    const int half = lane >> 4;                        // 0: lanes 0-15, 1: lanes 16-31
    const int l16  = lane & 15;
    const int koff = half << 2;                        // 0 or 4

    const float* wq0 = W + (size_t)(n0      + l16) * K + koff;  // B tile 0
    const float* wq1 = W + (size_t)(n0 + 16 + l16) * K + koff;  // B tile 1
    const float* xp0 = X + (size_t)( 0 + l16) * K + koff;       // A rows
    const float* xp1 = X + (size_t)(16 + l16) * K + koff;
    const float* xp2 = X + (size_t)(32 + l16) * K + koff;
    const float* xp3 = X + (size_t)(48 + l16) * K + koff;

    v8f c00 = {}, c01 = {}, c02 = {}, c03 = {};        // N tile 0, M tiles 0..3
    v8f c10 = {}, c11 = {}, c12 = {}, c13 = {};        // N tile 1

    // prologue: chunk 0
    v4f b0 = __builtin_nontemporal_load((const v4f*)wq0);
    v4f b1 = __builtin_nontemporal_load((const v4f*)wq1);
    v4f a0 = *(const v4f*)xp0;
    v4f a1 = *(const v4f*)xp1;
    v4f a2 = *(const v4f*)xp2;
    v4f a3 = *(const v4f*)xp3;

    for (int k0 = 8; k0 < K; k0 += 8) {
        // issue next-chunk loads first (stay in flight during WMMAs)
        v4f bn0 = __builtin_nontemporal_load((const v4f*)(wq0 + k0));
        v4f bn1 = __builtin_nontemporal_load((const v4f*)(wq1 + k0));
        v4f an0 = *(const v4f*)(xp0 + k0);
        v4f an1 = *(const v4f*)(xp1 + k0);
        v4f an2 = *(const v4f*)(xp2 + k0);
        v4f an3 = *(const v4f*)(xp3 + k0);

        const v2f b0l = lo2(b0), b0h = hi2(b0);
        const v2f b1l = lo2(b1), b1h = hi2(b1);
        const v2f a0l = lo2(a0), a0h = hi2(a0);
        const v2f a1l = lo2(a1), a1h = hi2(a1);
        const v2f a2l = lo2(a2), a2h = hi2(a2);
        const v2f a3l = lo2(a3), a3h = hi2(a3);

        c00 = WMMA_F32(a0l, b0l, c00);
        c01 = WMMA_F32(a1l, b0l, c01);
        c02 = WMMA_F32(a2l, b0l, c02);
        c03 = WMMA_F32(a3l, b0l, c03);
        c10 = WMMA_F32(a0l, b1l, c10);
        c11 = WMMA_F32(a1l, b1l, c11);
        c12 = WMMA_F32(a2l, b1l, c12);
        c13 = WMMA_F32(a3l, b1l, c13);
        c00 = WMMA_F32(a0h, b0h, c00);
        c01 = WMMA_F32(a1h, b0h, c01);
        c02 = WMMA_F32(a2h, b0h, c02);
        c03 = WMMA_F32(a3h, b0h, c03);
        c10 = WMMA_F32(a0h, b1h, c10);
        c11 = WMMA_F32(a1h, b1h, c11);
        c12 = WMMA_F32(a2h, b1h, c12);
        c13 = WMMA_F32(a3h, b1h, c13);

        b0 = bn0; b1 = bn1; a0 = an0; a1 = an1; a2 = an2; a3 = an3;
    }
    { // epilogue: last chunk
        const v2f b0l = lo2(b0), b0h = hi2(b0);
        const v2f b1l = lo2(b1), b1h = hi2(b1);
        const v2f a0l = lo2(a0), a0h = hi2(a0);
        const v2f a1l = lo2(a1), a1h = hi2(a1);
        const v2f a2l = lo2(a2), a2h = hi2(a2);
        const v2f a3l = lo2(a3), a3h = hi2(a3);
        c00 = WMMA_F32(a0l, b0l, c00);
        c01 = WMMA_F32(a1l, b0l, c01);
        c02 = WMMA_F32(a2l, b0l, c02);
        c03 = WMMA_F32(a3l, b0l, c03);
        c10 = WMMA_F32(a0l, b1l, c10);
        c11 = WMMA_F32(a1l, b1l, c11);
        c12 = WMMA_F32(a2l, b1l, c12);
        c13 = WMMA_F32(a3l, b1l, c13);
        c00 = WMMA_F32(a0h, b0h, c00);
        c01 = WMMA_F32(a1h, b0h, c01);
        c02 = WMMA_F32(a2h, b0h, c02);
        c03 = WMMA_F32(a3h, b0h, c03);
        c10 = WMMA_F32(a0h, b1h, c10);
        c11 = WMMA_F32(a1h, b1h, c11);
        c12 = WMMA_F32(a2h, b1h, c12);
        c13 = WMMA_F32(a3h, b1h, c13);
    }

    const float bv0 = bias ? bias[n0      + l16] : 0.0f;
    const float bv1 = bias ? bias[n0 + 16 + l16] : 0.0f;
    // D layout: lane covers col; VGPR v holds row m = mt*16 + v + 8*half
    #pragma unroll
    for (int v = 0; v < 8; ++v) {
        const int mrow = v + (half << 3);
        D[(size_t)( 0 + mrow) * N + n0      + l16] = c00[v] + bv0;
        D[(size_t)(16 + mrow) * N + n0      + l16] = c01[v] + bv0;
        D[(size_t)(32 + mrow) * N + n0      + l16] = c02[v] + bv0;
        D[(size_t)(48 + mrow) * N + n0      + l16] = c03[v] + bv0;
        D[(size_t)( 0 + mrow) * N + n0 + 16 + l16] = c10[v] + bv1;
        D[(size_t)(16 + mrow) * N + n0 + 16 + l16] = c11[v] + bv1;
        D[(size_t)(32 + mrow) * N + n0 + 16 + l16] = c12[v] + bv1;
        D[(size_t)(48 + mrow) * N + n0 + 16 + l16] = c13[v] + bv1;
    }
}

// ---------------------------------------------------------------------------
// Attention + embedding gather.  One block per batch row, 128 threads
// (one per source position).  Writes x = [emb | context], y[:,512:] = context.
// ---------------------------------------------------------------------------
__global__ void attn_embed_kernel(const int*   __restrict__ input_token,
                                  const float* __restrict__ hidden,     // [64,512]
                                  const float* __restrict__ enc_out,    // [64,128,512]
                                  const int*   __restrict__ src_mask,   // [64,128]
                                  const float* __restrict__ emb_table,  // [V,512]
                                  float* __restrict__ x,                // [64,1024]
                                  float* __restrict__ y) {              // [64,1024]
    const int b = blockIdx.x;
    const int t = threadIdx.x;                 // 0..127
    __shared__ float q[HID];
    __shared__ float sc[SLEN];
    __shared__ float red[SLEN];

    for (int h = t; h < HID; h += 128) q[h] = hidden[(size_t)b * HID + h];
    __syncthreads();

    // scores[b, t] = dot(enc_out[b, t, :], q)
    const float* erow = enc_out + ((size_t)b * SLEN + t) * HID;
    float s = 0.0f;
    for (int h = 0; h < HID; ++h) s = fmaf(erow[h], q[h], s);
    if (src_mask[b * SLEN + t] == 0) s = -1e9f;
    sc[t]  = s;
    red[t] = s;
    __syncthreads();

    // max reduction
    for (int off = 64; off > 0; off >>= 1) {
        if (t < off) red[t] = fmaxf(red[t], red[t + off]);
        __syncthreads();
    }
    const float mx = red[0];
    __syncthreads();

    const float e = __expf(sc[t] - mx);
    sc[t]  = e;
    red[t] = e;
    __syncthreads();

    // sum reduction
    for (int off = 64; off > 0; off >>= 1) {
        if (t < off) red[t] += red[t + off];
        __syncthreads();
    }
    const float inv = 1.0f / red[0];
    __syncthreads();
    sc[t] *= inv;
    __syncthreads();

    // context[b, h] = sum_s w[s] * enc_out[b, s, h]  (coalesced over h)
    for (int i = 0; i < 4; ++i) {
        const int h = t + 128 * i;
        float acc = 0.0f;
        for (int s2 = 0; s2 < SLEN; ++s2)
            acc = fmaf(sc[s2], enc_out[((size_t)b * SLEN + s2) * HID + h], acc);
        x[(size_t)b * 1024 + EMB + h] = acc;   // x = [emb | context]
        y[(size_t)b * 1024 + HID + h] = acc;   // y = [h_new | context]
    }

    // embedding gather
    const int tok = input_token[b];
    for (int h = t; h < EMB; h += 128)
        x[(size_t)b * 1024 + h] = emb_table[(size_t)tok * EMB + h];
}

// ---------------------------------------------------------------------------
// GRU pointwise gates (PyTorch order r, z, n); writes h_new into y[:, :512]
// and into the h_new output slot.
// ---------------------------------------------------------------------------
__global__ void gru_gate_kernel(const float* __restrict__ gx,     // [64,1536]
                                const float* __restrict__ gh,     // [64,1536]
                                const float* __restrict__ hidden, // [64,512]
                                float* __restrict__ y,            // [64,1024]
                                float* __restrict__ hout) {       // [64,512]
    const int idx = blockIdx.x * blockDim.x + threadIdx.x;
    if (idx >= BATCH * HID) return;
    const int b = idx >> 9;
    const int h = idx & (HID - 1);
    const size_t g = (size_t)b * 3 * HID + h;
    const float xr = gx[g], xz = gx[g + HID], xn = gx[g + 2 * HID];
    const float hr = gh[g], hz = gh[g + HID], hn = gh[g + 2 * HID];
    const float r = 1.0f / (1.0f + __expf(-(xr + hr)));
    const float z = 1.0f / (1.0f + __expf(-(xz + hz)));
    const float n = tanhf(xn + r * hn);
    const float hnew = (1.0f - z) * n + z * hidden[idx];
    y[(size_t)b * 1024 + h] = hnew;
    hout[idx] = hnew;
}

extern "C" void kernel_launch(void* const* d_in, const int* in_sizes, int n_in,
                              void* d_out, int out_size, void* d_ws, size_t ws_size,
                              hipStream_t stream) {
    const int*   input_token = (const int*)  d_in[0];
    const float* hidden      = (const float*)d_in[1];   // [1,64,512]
    const float* enc_out     = (const float*)d_in[2];   // [64,128,512]
    const int*   src_mask    = (const int*)  d_in[3];   // [64,128]
    const float* emb_table   = (const float*)d_in[4];   // [32000,512]
    const float* W_ih        = (const float*)d_in[5];   // [1536,1024]
    const float* W_hh        = (const float*)d_in[6];   // [1536,512]
    const float* b_ih        = (const float*)d_in[7];   // [1536]
    const float* b_hh        = (const float*)d_in[8];   // [1536]
    const float* W_out       = (const float*)d_in[9];   // [32000,1024]
    const float* b_out       = (const float*)d_in[10];  // [32000]

    float* logits = (float*)d_out;                      // [64,32000]
    float* hout   = logits + (size_t)BATCH * VOCAB;     // [1,64,512]

    float* ws = (float*)d_ws;
    float* x  = ws;                                     // [64,1024]
    float* y  = x  + BATCH * 1024;                      // [64,1024]
    float* gx = y  + BATCH * 1024;                      // [64,1536]
    float* gh = gx + BATCH * 3 * HID;                   // [64,1536]

    // 1) attention + embedding -> x, y[:,512:]
    attn_embed_kernel<<<BATCH, 128, 0, stream>>>(input_token, hidden, enc_out,
                                                 src_mask, emb_table, x, y);

    // 2) gx = x @ W_ih^T + b_ih   (N=1536, K=1024) : 48 wave-units, 4 waves/block
    wmma_gemm_xwT<<<12, 128, 0, stream>>>(x, W_ih, b_ih, gx, 3 * HID, EMB + HID);

    // 3) gh = h @ W_hh^T + b_hh   (N=1536, K=512)
    wmma_gemm_xwT<<<12, 128, 0, stream>>>(hidden, W_hh, b_hh, gh, 3 * HID, HID);

    // 4) gates -> h_new into y[:, :512] and hout
    gru_gate_kernel<<<(BATCH * HID + 255) / 256, 256, 0, stream>>>(gx, gh, hidden, y, hout);

    // 5) logits = y @ W_out^T + b_out  (N=32000, K=1024) : 1000 wave-units
    wmma_gemm_xwT<<<250, 128, 0, stream>>>(y, W_out, b_out, logits, VOCAB, EMB + HID);
}